// ConvertParamsLayerFrom0_16277926052103
// MI455X (gfx1250) — compile-verified
//
#include <hip/hip_runtime.h>
#include <hip/hip_bf16.h>

// Problem constants from the reference
#define BATCH 2048
#define NH    128
#define NV    256

typedef __attribute__((ext_vector_type(2))) float v2f;
typedef __attribute__((ext_vector_type(8))) float v8f;

// One block = 256 threads (8 wave32) handles (batch, vhalf). Each wave owns a
// 16-wide column stripe of the 128x256 per-batch matrix:
//   - streams wt1 once (NT loads), stores scaled copy to wt2 (NT stores)
//   - contracts the h dimension with v_wmma_f32_16x16x4_f32 at full f32
//     precision (A = s-vector replicated across M rows -> all D rows equal)
__global__ void __launch_bounds__(256)
convert_params_fused_kernel(const float* __restrict__ b1,
                            const float* __restrict__ wt1,
                            const float* __restrict__ muh2,
                            const float* __restrict__ cov,
                            float* __restrict__ b2,
                            float* __restrict__ wt2) {
    const int batch = blockIdx.x >> 1;   // 0..2047
    const int vhalf = blockIdx.x & 1;    // 0..1
    const int tid   = threadIdx.x;
    const int wave  = tid >> 5;          // 0..7
    const int lane  = tid & 31;          // 0..31

    __shared__ float s_s[NH];    // rsqrt(cov)*muh2  (WMMA A vector)
    __shared__ float s_inv[NH];  // rsqrt(cov)       (row scale for wt2)

    if (tid < NH) {
        const float c  = cov[batch * NH + tid];
        const float is = __frsqrt_rn(c);
        s_inv[tid] = is;
        s_s[tid]   = is * muh2[batch * NH + tid];
    }
    __syncthreads();

    // This wave's 16-column stripe
    const int v0 = (vhalf * 8 + wave) * 16;

    const long mbase = (long)batch * (NH * NV);
    const float* __restrict__ wsrc = wt1 + mbase;
    float* __restrict__       wdst = wt2 + mbase;

    // WMMA f32 16x16x4 lane layout (mirrored for A and B):
    //   VGPR j, lanes 0-15  -> K = j     ; lanes 16-31 -> K = j + 2
    const int rowoff = (lane >= 16) ? 2 : 0;
    const int col    = v0 + (lane & 15);

    v8f acc = {};  // D rows all identical; row 0 = result for 16 v's

    #pragma unroll 4
    for (int h0 = 0; h0 < NH; h0 += 4) {
        const int r0 = h0 + rowoff;      // row for VGPR0 slot
        const int r1 = h0 + 1 + rowoff;  // row for VGPR1 slot

        // B tile: two rows per lane, 64B-contiguous per 16-lane half
        const float w0 = __builtin_nontemporal_load(wsrc + (long)r0 * NV + col);
        const float w1 = __builtin_nontemporal_load(wsrc + (long)r1 * NV + col);

        // Fused wt2 = wt1 * rsqrt(cov) row-scaling (same registers, NT store)
        __builtin_nontemporal_store(w0 * s_inv[r0], wdst + (long)r0 * NV + col);
        __builtin_nontemporal_store(w1 * s_inv[r1], wdst + (long)r1 * NV + col);

        // A vector (replicated across all 16 M rows): s[h] in matching K slots
        v2f A; A.x = s_s[r0]; A.y = s_s[r1];
        v2f Bm; Bm.x = w0;    Bm.y = w1;

        // acc = A(16x4) x B(4x16) + acc  — f32 WMMA, K=4 contraction over h
        acc = __builtin_amdgcn_wmma_f32_16x16x4_f32(
            /*neg_a=*/false, A, /*neg_b=*/false, Bm,
            /*c_mod=*/(short)0, acc, /*reuse_a=*/false, /*reuse_b=*/false);
    }

    // D row M=0 lives in acc[0], lanes 0-15 (N = lane)
    if (lane < 16) {
        const int v = v0 + lane;
        b2[batch * NV + v] = b1[batch * NV + v] - acc[0];
    }
}

extern "C" void kernel_launch(void* const* d_in, const int* in_sizes, int n_in,
                              void* d_out, int out_size, void* d_ws, size_t ws_size,
                              hipStream_t stream) {
    (void)in_sizes; (void)n_in; (void)out_size; (void)d_ws; (void)ws_size;

    const float* b1   = (const float*)d_in[0];  // [B, NV]
    const float* wt1  = (const float*)d_in[1];  // [B, NH, NV]
    const float* muh2 = (const float*)d_in[2];  // [B, NH]
    const float* cov  = (const float*)d_in[3];  // [B, NH]

    float* b2  = (float*)d_out;                 // [B, NV]   (first output)
    float* wt2 = b2 + (long)BATCH * NV;         // [B, NH, NV] (second output)

    dim3 grid(BATCH * 2);
    dim3 block(256);
    convert_params_fused_kernel<<<grid, block, 0, stream>>>(b1, wt1, muh2, cov, b2, wt2);
}